// Embedder_37512244364079
// MI455X (gfx1250) — compile-verified
//
#include <hip/hip_runtime.h>
#include <hip/hip_bf16.h>
#include <stdint.h>

// ---- problem constants (match reference) ----
#define BATCH     8
#define NPTS      1024
#define SEEDS_PER 128          // S = N * RATIO
#define NSEEDS    (BATCH*SEEDS_PER)   // 1024
#define KNBR      32
#define H1        128
#define F1        256
#define H2        512
#define EMBD      384

typedef float v2f __attribute__((ext_vector_type(2)));
typedef float v8f __attribute__((ext_vector_type(8)));

// f32 WMMA: D[16x16] += A[16x4] x B[4x16], full fp32 precision (CDNA5).
__device__ __forceinline__ v8f wmma4(v2f a, v2f b, v8f c) {
    return __builtin_amdgcn_wmma_f32_16x16x4_f32(
        /*neg_a=*/false, a, /*neg_b=*/false, b,
        /*c_mod=*/(short)0, c, /*reuse_a=*/false, /*reuse_b=*/false);
}

__device__ __forceinline__ v8f v8zero() {
    v8f z;
#pragma unroll
    for (int i = 0; i < 8; ++i) z[i] = 0.0f;
    return z;
}

// ======================= 1) Farthest Point Sampling =======================
// One block per cloud. 4 points per thread held in registers. Argmax via
// packed-key LDS atomicMax; key low bits (1023-idx) => ties pick lowest idx.
__global__ __launch_bounds__(256) void fps_kernel(const float* __restrict__ pos,
                                                  float* __restrict__ seeds) {
    const int b = blockIdx.x;
    const int t = threadIdx.x;
    const float* pb = pos + (size_t)b * NPTS * 3;

    float px[4], py[4], pz[4], mind[4];
#pragma unroll
    for (int i = 0; i < 4; ++i) {
        int n = t * 4 + i;
        px[i] = pb[n * 3 + 0];
        py[i] = pb[n * 3 + 1];
        pz[i] = pb[n * 3 + 2];
        mind[i] = 3.4e38f;
    }

    __shared__ unsigned long long s_key;
    float lx = pb[0], ly = pb[1], lz = pb[2];   // start index 0 (deterministic)

    for (int s = 0; s < SEEDS_PER; ++s) {
        if (t == 0) {
            float* sp = seeds + (size_t)(b * SEEDS_PER + s) * 3;
            sp[0] = lx; sp[1] = ly; sp[2] = lz;
            s_key = 0ull;
        }
        __syncthreads();
        unsigned long long lk = 0ull;
#pragma unroll
        for (int i = 0; i < 4; ++i) {
            float dx = px[i] - lx, dy = py[i] - ly, dz = pz[i] - lz;
            float d = dx * dx + dy * dy + dz * dz;
            mind[i] = fminf(mind[i], d);
            unsigned long long key =
                ((unsigned long long)__float_as_uint(mind[i]) << 32) |
                (unsigned)(1023 - (t * 4 + i));
            lk = (key > lk) ? key : lk;
        }
        atomicMax(&s_key, lk);
        __syncthreads();
        int n = 1023 - (int)(unsigned)(s_key & 0xffffffffull);
        lx = pb[n * 3 + 0];
        ly = pb[n * 3 + 1];
        lz = pb[n * 3 + 2];
        __syncthreads();   // protect s_key against next-iter reset
    }
}

// ======================= 2) kNN (K=32 of 1024) ============================
// One block per seed. 4 distances per thread in registers; 32 packed-key
// atomicMin extractions (tie -> lowest index, matching jax top_k).
__global__ __launch_bounds__(256) void knn_kernel(const float* __restrict__ pos,
                                                  const float* __restrict__ seeds,
                                                  int* __restrict__ nbr) {
    const int bs = blockIdx.x;
    const int b  = bs >> 7;      // / SEEDS_PER
    const int t  = threadIdx.x;
    const float* pb = pos + (size_t)b * NPTS * 3;
    const float sx = seeds[bs * 3 + 0];
    const float sy = seeds[bs * 3 + 1];
    const float sz = seeds[bs * 3 + 2];

    unsigned long long key[4];
#pragma unroll
    for (int i = 0; i < 4; ++i) {
        int n = t * 4 + i;
        float dx = pb[n * 3 + 0] - sx;
        float dy = pb[n * 3 + 1] - sy;
        float dz = pb[n * 3 + 2] - sz;
        float d = dx * dx + dy * dy + dz * dz;
        key[i] = ((unsigned long long)__float_as_uint(d) << 32) | (unsigned)n;
    }

    __shared__ unsigned long long s_key;
    for (int k = 0; k < KNBR; ++k) {
        if (t == 0) s_key = ~0ull;
        __syncthreads();
        unsigned long long lk = key[0];
#pragma unroll
        for (int i = 1; i < 4; ++i) lk = (key[i] < lk) ? key[i] : lk;
        atomicMin(&s_key, lk);
        __syncthreads();
        unsigned long long w = s_key;
        int n = (int)(unsigned)(w & 0xffffffffull);
        if (t == 0) nbr[bs * KNBR + k] = n;
        if ((n >> 2) == t) key[n & 3] = ~0ull;   // remove winner
        __syncthreads();
    }
}

// ======================= 3) MLP1: rel -> 128 -> 256 =======================
// 64 rows (2 seeds) per block. Layer 3->128 in VALU (K=3 too small for WMMA),
// layer 128->256 as f32 WMMA GEMM with A in LDS, B streamed from L2.
#define HSTR 130   // padded stride (even for b64 LDS loads, breaks bank conflicts)
__global__ __launch_bounds__(256) void mlp1_kernel(
    const float* __restrict__ pos, const float* __restrict__ seeds,
    const int* __restrict__ nbr,
    const float* __restrict__ W1a, const float* __restrict__ b1a,
    const float* __restrict__ W1b, const float* __restrict__ b1b,
    float* __restrict__ f_ws) {
    __shared__ float s_rel[64][3];
    __shared__ float s_w1a[3 * H1];
    __shared__ float s_b1a[H1];
    __shared__ float s_hid[64 * HSTR];

    const int t   = threadIdx.x;
    const int blk = blockIdx.x;   // 512 blocks x 64 rows

    for (int j = t; j < 3 * H1; j += 256) s_w1a[j] = W1a[j];
    if (t < H1) s_b1a[t] = b1a[t];
    if (t < 64) {
        int row = blk * 64 + t;          // global neighbor-row
        int bs  = row >> 5;              // seed id
        int kk  = row & 31;
        int n   = nbr[bs * KNBR + kk];
        int b   = bs >> 7;
        const float* pp = pos + (size_t)(b * NPTS + n) * 3;
        const float* sp = seeds + (size_t)bs * 3;
        s_rel[t][0] = pp[0] - sp[0];
        s_rel[t][1] = pp[1] - sp[1];
        s_rel[t][2] = pp[2] - sp[2];
    }
    __syncthreads();

    // hidden = relu(rel @ W1a + b1a)   [64][128]
    for (int e = t; e < 64 * H1; e += 256) {
        int r = e >> 7, c = e & 127;
        float h = s_b1a[c] + s_rel[r][0] * s_w1a[c]
                           + s_rel[r][1] * s_w1a[H1 + c]
                           + s_rel[r][2] * s_w1a[2 * H1 + c];
        s_hid[r * HSTR + c] = fmaxf(h, 0.0f);
    }
    __syncthreads();

    // f = hidden @ W1b + b1b   [64][256] -> global workspace
    const int wave = t >> 5, lane = t & 31;
    const int lm = lane & 15, halo = lane >> 4;
    for (int nt = wave; nt < 16; nt += 8) {
        const int n0 = nt * 16 + lm;
        v8f acc[4];
#pragma unroll
        for (int mt = 0; mt < 4; ++mt) acc[mt] = v8zero();
        for (int k = 0; k < H1; k += 4) {
            int kk = k + 2 * halo;
            v2f bf;
            bf.x = W1b[(size_t)kk * F1 + n0];
            bf.y = W1b[(size_t)(kk + 1) * F1 + n0];
#pragma unroll
            for (int mt = 0; mt < 4; ++mt) {
                int m = mt * 16 + lm;
                v2f af;
                af.x = s_hid[m * HSTR + kk];
                af.y = s_hid[m * HSTR + kk + 1];
                acc[mt] = wmma4(af, bf, acc[mt]);
            }
        }
        float bias = b1b[n0];
#pragma unroll
        for (int mt = 0; mt < 4; ++mt) {
#pragma unroll
            for (int v = 0; v < 8; ++v) {
                int row = blk * 64 + mt * 16 + v + 8 * halo;
                f_ws[(size_t)row * F1 + n0] = acc[mt][v] + bias;
            }
        }
    }
}

// ============ 4) MLP2 layer A: a = relu([g|f] @ W2a + b2a) ================
// One seed per block; g (column max over K=32) is identical for all 32 rows,
// so the 512-wide concat input is synthesized on the fly: k<256 -> g, else f.
#define FS 258
__global__ __launch_bounds__(256) void mlp2a_kernel(
    const float* __restrict__ f_ws, const float* __restrict__ W2a,
    const float* __restrict__ b2a, float* __restrict__ a_ws) {
    __shared__ float s_f[32 * FS];
    __shared__ float s_g[F1];
    const int bs = blockIdx.x;
    const int t  = threadIdx.x;
    const float* fb = f_ws + (size_t)bs * KNBR * F1;

    for (int e = t; e < 32 * F1; e += 256) {
        int r = e >> 8, c = e & 255;
        s_f[r * FS + c] = fb[e];
    }
    __syncthreads();
    {   // g[c] = max_r f[r][c]
        float m = s_f[t];
        for (int r = 1; r < 32; ++r) m = fmaxf(m, s_f[r * FS + t]);
        s_g[t] = m;
    }
    __syncthreads();

    const int wave = t >> 5, lane = t & 31;
    const int lm = lane & 15, halo = lane >> 4;
    for (int nt = wave; nt < 32; nt += 8) {
        const int n0 = nt * 16 + lm;
        v8f acc0 = v8zero(), acc1 = v8zero();
        for (int k = 0; k < H2; k += 4) {
            int kk = k + 2 * halo;
            v2f bf;
            bf.x = W2a[(size_t)kk * H2 + n0];
            bf.y = W2a[(size_t)(kk + 1) * H2 + n0];
            v2f a0, a1;
            if (k < F1) {               // g-half of concat (same for all rows)
                a0.x = s_g[kk]; a0.y = s_g[kk + 1];
                a1 = a0;
            } else {                    // f-half
                int kc = kk - F1;
                a0.x = s_f[lm * FS + kc];        a0.y = s_f[lm * FS + kc + 1];
                a1.x = s_f[(16 + lm) * FS + kc]; a1.y = s_f[(16 + lm) * FS + kc + 1];
            }
            acc0 = wmma4(a0, bf, acc0);
            acc1 = wmma4(a1, bf, acc1);
        }
        float bias = b2a[n0];
#pragma unroll
        for (int v = 0; v < 8; ++v) {
            int r0 = v + 8 * halo;
            a_ws[((size_t)bs * KNBR + r0)      * H2 + n0] = fmaxf(acc0[v] + bias, 0.0f);
            a_ws[((size_t)bs * KNBR + 16 + r0) * H2 + n0] = fmaxf(acc1[v] + bias, 0.0f);
        }
    }
}

// ============ 5) MLP2 layer B + max-pool: out = max_k(a @ W2b) + b2b ======
// One seed per block; max over the 32 neighbor rows done in-register per
// fragment + cross-half shuffle, so no atomics and no out-buffer init needed.
__global__ __launch_bounds__(256) void mlp2b_kernel(
    const float* __restrict__ a_ws, const float* __restrict__ W2b,
    const float* __restrict__ b2b, float* __restrict__ out) {
    __shared__ float s_a[32 * H2];   // exactly 64 KB
    const int bs = blockIdx.x;
    const int t  = threadIdx.x;
    const float* ab = a_ws + (size_t)bs * KNBR * H2;
    for (int e = t; e < 32 * H2; e += 256) s_a[e] = ab[e];
    __syncthreads();

    const int wave = t >> 5, lane = t & 31;
    const int lm = lane & 15, halo = lane >> 4;
    for (int nt = wave; nt < 24; nt += 8) {
        const int n0 = nt * 16 + lm;
        v8f acc0 = v8zero(), acc1 = v8zero();
        for (int k = 0; k < H2; k += 4) {
            int kk = k + 2 * halo;
            v2f bf;
            bf.x = W2b[(size_t)kk * EMBD + n0];
            bf.y = W2b[(size_t)(kk + 1) * EMBD + n0];
            v2f a0, a1;
            a0.x = s_a[lm * H2 + kk];        a0.y = s_a[lm * H2 + kk + 1];
            a1.x = s_a[(16 + lm) * H2 + kk]; a1.y = s_a[(16 + lm) * H2 + kk + 1];
            acc0 = wmma4(a0, bf, acc0);
            acc1 = wmma4(a1, bf, acc1);
        }
        // rows covered by this lane: halo half of both 16-row M-tiles
        float p = -3.4e38f;
#pragma unroll
        for (int v = 0; v < 8; ++v) p = fmaxf(p, fmaxf(acc0[v], acc1[v]));
        // combine the two lane halves (rows v+0..7 vs v+8..15)
        float q = fmaxf(p, __shfl_xor(p, 16, 32));
        if (lane < 16) out[(size_t)bs * EMBD + n0] = q + b2b[n0];
    }
}

// =========================== launcher =====================================
extern "C" void kernel_launch(void* const* d_in, const int* in_sizes, int n_in,
                              void* d_out, int out_size, void* d_ws, size_t ws_size,
                              hipStream_t stream) {
    const float* pos  = (const float*)d_in[0];
    // d_in[1] = batch (contiguous, unused)
    const float* W1a  = (const float*)d_in[2];
    const float* b1a  = (const float*)d_in[3];
    const float* W1b  = (const float*)d_in[4];
    const float* b1b  = (const float*)d_in[5];
    const float* W2a  = (const float*)d_in[6];
    const float* b2a  = (const float*)d_in[7];
    const float* W2b  = (const float*)d_in[8];
    const float* b2b  = (const float*)d_in[9];
    float* out = (float*)d_out;

    char* ws = (char*)d_ws;
    float* seeds = (float*)ws;                                  //  12 KB
    int*   nbr   = (int*)(ws + 12288);                          // 128 KB
    float* f_ws  = (float*)(ws + 143360);                       //  32 MB
    float* a_ws  = (float*)(ws + 143360 + (size_t)32768 * F1 * 4); // 64 MB

    fps_kernel  <<<BATCH,  256, 0, stream>>>(pos, seeds);
    knn_kernel  <<<NSEEDS, 256, 0, stream>>>(pos, seeds, nbr);
    mlp1_kernel <<<512,    256, 0, stream>>>(pos, seeds, nbr, W1a, b1a, W1b, b1b, f_ws);
    mlp2a_kernel<<<NSEEDS, 256, 0, stream>>>(f_ws, W2a, b2a, a_ws);
    mlp2b_kernel<<<NSEEDS, 256, 0, stream>>>(a_ws, W2b, b2b, out);
}